// Attention_layer_41429254537559
// MI455X (gfx1250) — compile-verified
//
#include <hip/hip_runtime.h>
#include <cmath>

typedef __attribute__((ext_vector_type(2))) float v2f;
typedef __attribute__((ext_vector_type(8))) float v8f;

#define N_ 16
#define L_ 196
#define D_ 1024
#define T_ 32

// ---------------------------------------------------------------------------
// Kernel 1: img_score[n,l] = img[n,l,:] . w ; q_score[n,t] = qes[n,t,:] . w
// One wave32 per row; strided loads + xor-shuffle reduction.
// ---------------------------------------------------------------------------
__global__ void __launch_bounds__(256)
scores_kernel(const float* __restrict__ img, const float* __restrict__ qes,
              const float* __restrict__ w, float* __restrict__ img_score,
              float* __restrict__ q_score) {
  const int lane = threadIdx.x & 31;
  const int wave = threadIdx.x >> 5;
  const int row  = blockIdx.x * (blockDim.x >> 5) + wave;

  const int totalImg = N_ * L_;
  const float* src;
  float* dst;
  if (row < totalImg) {
    src = img + (size_t)row * D_;
    dst = img_score + row;
  } else {
    const int r2 = row - totalImg;
    if (r2 >= N_ * T_) return;
    src = qes + (size_t)r2 * D_;
    dst = q_score + r2;
  }

  float acc = 0.0f;
  #pragma unroll 8
  for (int d = lane; d < D_; d += 32) acc += src[d] * w[d];
  #pragma unroll
  for (int off = 16; off > 0; off >>= 1) acc += __shfl_xor(acc, off, 32);
  if (lane == 0) *dst = acc;
}

// ---------------------------------------------------------------------------
// Kernel 2: att[n,t,l] = softmax_l( img_score[n,l] + q_score[n,t] )
// One wave32 per (n,t) row; 7 strided elements per lane over L=196.
// ---------------------------------------------------------------------------
__global__ void __launch_bounds__(256)
softmax_kernel(const float* __restrict__ img_score,
               const float* __restrict__ q_score,
               float* __restrict__ att) {
  const int lane = threadIdx.x & 31;
  const int wave = threadIdx.x >> 5;
  const int row  = blockIdx.x * (blockDim.x >> 5) + wave;   // n*T + t
  if (row >= N_ * T_) return;
  const int n  = row / T_;
  const float qs = q_score[row];

  float vals[7];
  float m = -INFINITY;
  #pragma unroll
  for (int i = 0; i < 7; ++i) {
    const int l = lane + i * 32;
    const float v = (l < L_) ? (img_score[n * L_ + l] + qs) : -INFINITY;
    vals[i] = v;
    m = fmaxf(m, v);
  }
  #pragma unroll
  for (int off = 16; off > 0; off >>= 1) m = fmaxf(m, __shfl_xor(m, off, 32));

  float s = 0.0f;
  #pragma unroll
  for (int i = 0; i < 7; ++i) {
    const int l = lane + i * 32;
    const float e = (l < L_) ? __expf(vals[i] - m) : 0.0f;
    vals[i] = e;
    s += e;
  }
  #pragma unroll
  for (int off = 16; off > 0; off >>= 1) s += __shfl_xor(s, off, 32);
  const float inv = 1.0f / s;

  #pragma unroll
  for (int i = 0; i < 7; ++i) {
    const int l = lane + i * 32;
    if (l < L_) att[(size_t)row * L_ + l] = vals[i] * inv;
  }
}

// ---------------------------------------------------------------------------
// Kernel 3: out[n,t,:] = qes[n,t,:] + att[n,t,:] @ img[n,:,:]
// FP32 WMMA: D = A(16x4) x B(4x16) + C(16x16), K stepped over L=196 (49 it).
// Each wave: one 16-wide D tile, both 16-row T tiles (T=32).
// A layout: lane = M + 16*(K>=2); VGPR0 holds K∈{0,2}, VGPR1 holds K∈{1,3}.
// B layout: lane = N + 16*(K>=2); VGPR0 holds K∈{0,2}, VGPR1 holds K∈{1,3}.
// C layout: VGPR r -> (M=r, lanes 0-15) / (M=r+8, lanes 16-31).
// ---------------------------------------------------------------------------
__global__ void __launch_bounds__(256)
wmma_av_kernel(const float* __restrict__ att, const float* __restrict__ img,
               const float* __restrict__ qes, float* __restrict__ out) {
  const int lane   = threadIdx.x & 31;
  const int wave   = threadIdx.x >> 5;              // 0..7
  const int n      = blockIdx.y;
  const int d_base = blockIdx.x * 128 + wave * 16;  // 16-col D tile
  const int half   = lane >> 4;                     // 0 or 1
  const int nl     = lane & 15;

  const float* __restrict__ attn = att + (size_t)n * T_ * L_;
  const float* __restrict__ imgn = img + (size_t)n * L_ * D_;

  v8f c0 = {};  // T rows 0..15
  v8f c1 = {};  // T rows 16..31

  for (int k0 = 0; k0 < L_; k0 += 4) {
    const int kA = k0 + (half << 1);  // this lane's even K (x comp); +1 is y

    v2f a0, a1, b;
    // A fragments (att rows = t), pairs are contiguous in L
    const float* arow0 = attn + (size_t)nl * L_ + kA;
    const float* arow1 = attn + (size_t)(nl + 16) * L_ + kA;
    a0.x = arow0[0]; a0.y = arow0[1];
    a1.x = arow1[0]; a1.y = arow1[1];
    // B fragment (img rows = l, cols = d)
    const float* brow = imgn + (size_t)kA * D_ + d_base + nl;
    b.x = brow[0];
    b.y = brow[D_];

    c0 = __builtin_amdgcn_wmma_f32_16x16x4_f32(false, a0, false, b,
                                               (short)0, c0, false, false);
    c1 = __builtin_amdgcn_wmma_f32_16x16x4_f32(false, a1, false, b,
                                               (short)0, c1, false, false);
  }

  // Fused epilogue: out = qes + C
  const size_t base = ((size_t)n * T_) * D_ + d_base + nl;
  #pragma unroll
  for (int r = 0; r < 8; ++r) {
    const int t0 = r + half * 8;          // rows of c0
    const size_t i0 = base + (size_t)t0 * D_;
    out[i0] = qes[i0] + c0[r];
    const int t1 = t0 + 16;               // rows of c1
    const size_t i1 = base + (size_t)t1 * D_;
    out[i1] = qes[i1] + c1[r];
  }
}

// ---------------------------------------------------------------------------
extern "C" void kernel_launch(void* const* d_in, const int* in_sizes, int n_in,
                              void* d_out, int out_size, void* d_ws, size_t ws_size,
                              hipStream_t stream) {
  const float* img = (const float*)d_in[0];  // [N, L, D]
  const float* qes = (const float*)d_in[1];  // [N, T, D]
  const float* w   = (const float*)d_in[2];  // [D]
  float* out = (float*)d_out;                // [N, T, D]

  // Workspace layout (floats): img_score | q_score | att
  float* ws        = (float*)d_ws;
  float* img_score = ws;                         // N*L     = 3136
  float* q_score   = img_score + N_ * L_;        // N*T     = 512
  float* att       = q_score + N_ * T_;          // N*T*L   = 100352

  // Kernel 1: 3648 rows, 8 waves/block
  {
    const int rows = N_ * L_ + N_ * T_;
    const int blocks = (rows + 7) / 8;
    scores_kernel<<<blocks, 256, 0, stream>>>(img, qes, w, img_score, q_score);
  }
  // Kernel 2: 512 rows, 8 waves/block
  {
    const int rows = N_ * T_;
    const int blocks = (rows + 7) / 8;
    softmax_kernel<<<blocks, 256, 0, stream>>>(img_score, q_score, att);
  }
  // Kernel 3: grid (D/128, N), 8 waves/block, 2 WMMA tiles per wave
  {
    dim3 grid(D_ / 128, N_);
    wmma_av_kernel<<<grid, 256, 0, stream>>>(att, img, qes, out);
  }
}